// ALISTA_64948495450200
// MI455X (gfx1250) — compile-verified
//
#include <hip/hip_runtime.h>
#include <math.h>

typedef __attribute__((ext_vector_type(2))) float v2f;
typedef __attribute__((ext_vector_type(8))) float v8f;

#define BATCH 4096
#define MDIM  512
#define NDIM  2048
#define NITER 16

#define BM 128      // batch rows per block
#define BN 64       // output cols per block
#define KT 32       // K-stage depth
#define KS 36       // LDS row stride (float4-aligned rows, conflict-free WMMA reads)

// ---------------------------------------------------------------------------
// Register-staged tile movement (for double buffering)
// ---------------------------------------------------------------------------

// NR rows of 32 contiguous floats -> NR/32 float4 regs per thread (256 thr).
template <int NR>
__device__ __forceinline__ void ld_rows32_regs(const float* __restrict__ g, int ldg,
                                               int tid, float4 (&regs)[NR / 32]) {
#pragma unroll
  for (int j = 0; j < NR / 32; ++j) {
    const int i   = tid + j * 256;
    const int row = i >> 3;
    const int q   = (i & 7) << 2;
    regs[j] = *(const float4*)(g + (size_t)row * ldg + q);
  }
}

template <int NR>
__device__ __forceinline__ void st_rows32_lds(float* lds, int tid,
                                              const float4 (&regs)[NR / 32]) {
#pragma unroll
  for (int j = 0; j < NR / 32; ++j) {
    const int i   = tid + j * 256;
    const int row = i >> 3;
    const int q   = (i & 7) << 2;
    *(float4*)&lds[row * KS + q] = regs[j];
  }
}

// One prefetch per 128B row (rows of 32 floats) -> global_prefetch_b8.
template <int NR>
__device__ __forceinline__ void prefetch_rows32(const float* __restrict__ g, int ldg,
                                                int tid) {
  for (int i = tid; i < NR; i += 256)
    __builtin_prefetch(g + (size_t)i * ldg, 0, 0);
}

// W tile [KT x BN] read coalesced along n; 8 scalar regs per thread.
__device__ __forceinline__ void ld_wtile_regs(const float* __restrict__ W, int ldw,
                                              int n0, int ks, int tid, float (&regs)[8]) {
#pragma unroll
  for (int j = 0; j < 8; ++j) {
    const int i   = tid + j * 256;
    const int k   = i >> 6;
    const int col = i & 63;
    regs[j] = W[(size_t)(ks + k) * ldw + n0 + col];
  }
}

// Transpose-scatter into Bs[col][k].
__device__ __forceinline__ void st_wtile_lds(float* Bs, int tid, const float (&regs)[8]) {
#pragma unroll
  for (int j = 0; j < 8; ++j) {
    const int i   = tid + j * 256;
    const int k   = i >> 6;
    const int col = i & 63;
    Bs[col * KS + k] = regs[j];
  }
}

// ---------------------------------------------------------------------------
// 32x32 wave tile of WMMA over a KT-deep LDS stage.
// As[row][k] = A-operand (M x K), Bs[col][k] = B-transposed (N x K).
// ---------------------------------------------------------------------------
__device__ __forceinline__ void wmma_tile32(const float* As, const float* Bs,
                                            int wm, int wn, int rc, int kb,
                                            v8f acc[2][2]) {
#pragma unroll
  for (int kk = 0; kk < KT; kk += 4) {
    v2f a0 = *(const v2f*)&As[(wm      + rc) * KS + kk + kb];
    v2f a1 = *(const v2f*)&As[(wm + 16 + rc) * KS + kk + kb];
    v2f b0 = *(const v2f*)&Bs[(wn      + rc) * KS + kk + kb];
    v2f b1 = *(const v2f*)&Bs[(wn + 16 + rc) * KS + kk + kb];
    acc[0][0] = __builtin_amdgcn_wmma_f32_16x16x4_f32(false, a0, false, b0,
                                                      (short)0, acc[0][0], false, false);
    acc[0][1] = __builtin_amdgcn_wmma_f32_16x16x4_f32(false, a0, false, b1,
                                                      (short)0, acc[0][1], false, false);
    acc[1][0] = __builtin_amdgcn_wmma_f32_16x16x4_f32(false, a1, false, b0,
                                                      (short)0, acc[1][0], false, false);
    acc[1][1] = __builtin_amdgcn_wmma_f32_16x16x4_f32(false, a1, false, b1,
                                                      (short)0, acc[1][1], false, false);
  }
}

// ---------------------------------------------------------------------------
// Kernel 1:  r[b,m] = sum_n d[b,n] * A[m,n]  -  y[b,m]     (haveD=0 -> r=-y)
// Double-buffered LDS pipeline, one barrier per K-stage.
// ---------------------------------------------------------------------------
__global__ __launch_bounds__(256) void alista_rstep(
    const float* __restrict__ dprev,   // [BATCH, NDIM] or unused
    const float* __restrict__ Amat,    // [MDIM, NDIM]
    const float* __restrict__ y,       // [BATCH, MDIM]
    float* __restrict__ r,             // [BATCH, MDIM]
    int haveD) {
  __shared__ float As[2][BM * KS];
  __shared__ float Bs[2][BN * KS];

  const int tid  = threadIdx.x;
  const int c0   = blockIdx.x * BN;    // m-column block
  const int b0   = blockIdx.y * BM;    // batch-row block
  const int lane = tid & 31;
  const int rc   = lane & 15;
  const int hi   = lane >> 4;
  const int kb   = hi << 1;
  const int wave = tid >> 5;
  const int wm   = (wave & 3) * 32;
  const int wn   = (wave >> 2) * 32;

  v8f acc[2][2] = {};

  if (haveD) {
    const float* gA = dprev + (size_t)b0 * NDIM;   // A-operand rows (streaming)
    const float* gB = Amat  + (size_t)c0 * NDIM;   // B-operand rows (A-matrix)

    float4 ra[BM / 32];
    float4 rb[BN / 32];

    // Prologue: stage 0 into buffer 0.
    ld_rows32_regs<BM>(gA, NDIM, tid, ra);
    ld_rows32_regs<BN>(gB, NDIM, tid, rb);
    st_rows32_lds<BM>(As[0], tid, ra);
    st_rows32_lds<BN>(Bs[0], tid, rb);
    __syncthreads();

    int p = 0;
    for (int ks = 0; ks < NDIM; ks += KT) {
      const int nxt = ks + KT;
      if (nxt < NDIM) {
        ld_rows32_regs<BM>(gA + nxt, NDIM, tid, ra);
        ld_rows32_regs<BN>(gB + nxt, NDIM, tid, rb);
        if (nxt + KT < NDIM) {
          prefetch_rows32<BM>(gA + nxt + KT, NDIM, tid);
          prefetch_rows32<BN>(gB + nxt + KT, NDIM, tid);
        }
      }
      wmma_tile32(As[p], Bs[p], wm, wn, rc, kb, acc);
      if (nxt < NDIM) {
        st_rows32_lds<BM>(As[p ^ 1], tid, ra);
        st_rows32_lds<BN>(Bs[p ^ 1], tid, rb);
      }
      __syncthreads();
      p ^= 1;
    }
  }

#pragma unroll
  for (int rt = 0; rt < 2; ++rt)
#pragma unroll
    for (int ct = 0; ct < 2; ++ct)
#pragma unroll
      for (int i = 0; i < 8; ++i) {
        const int row = b0 + wm + rt * 16 + hi * 8 + i;
        const int col = c0 + wn + ct * 16 + rc;
        const size_t idx = (size_t)row * MDIM + col;
        r[idx] = acc[rt][ct][i] - y[idx];
      }
}

// ---------------------------------------------------------------------------
// Kernel 2:  z[b,n] = d[b,n] - s * sum_m r[b,m] * W[m,n]
//            out[b,n] = sign(z) * max(|z| - thr, 0)
// ---------------------------------------------------------------------------
__global__ __launch_bounds__(256) void alista_zstep(
    const float* __restrict__ dprev,   // [BATCH, NDIM] or unused
    const float* __restrict__ Wmat,    // [MDIM, NDIM]
    const float* __restrict__ r,       // [BATCH, MDIM]
    float* __restrict__ out,           // [BATCH, NDIM] (slice t)
    const float* __restrict__ thr_p,   // &thr[t]
    const float* __restrict__ step_p,  // &step[t]
    int haveD) {
  __shared__ float As[2][BM * KS];
  __shared__ float Bs[2][BN * KS];

  const int tid  = threadIdx.x;
  const int n0   = blockIdx.x * BN;    // n-column block
  const int b0   = blockIdx.y * BM;    // batch-row block
  const int lane = tid & 31;
  const int rc   = lane & 15;
  const int hi   = lane >> 4;
  const int kb   = hi << 1;
  const int wave = tid >> 5;
  const int wm   = (wave & 3) * 32;
  const int wn   = (wave >> 2) * 32;

  v8f acc[2][2] = {};

  const float* gA = r + (size_t)b0 * MDIM;   // A-operand rows (streaming)

  float4 ra[BM / 32];
  float  rw[8];

  // Prologue: stage 0 into buffer 0.
  ld_rows32_regs<BM>(gA, MDIM, tid, ra);
  ld_wtile_regs(Wmat, NDIM, n0, 0, tid, rw);
  st_rows32_lds<BM>(As[0], tid, ra);
  st_wtile_lds(Bs[0], tid, rw);
  __syncthreads();

  int p = 0;
  for (int ks = 0; ks < MDIM; ks += KT) {
    const int nxt = ks + KT;
    if (nxt < MDIM) {
      ld_rows32_regs<BM>(gA + nxt, MDIM, tid, ra);
      ld_wtile_regs(Wmat, NDIM, n0, nxt, tid, rw);
      if (nxt + KT < MDIM)
        prefetch_rows32<BM>(gA + nxt + KT, MDIM, tid);
    }
    wmma_tile32(As[p], Bs[p], wm, wn, rc, kb, acc);
    if (nxt < MDIM) {
      st_rows32_lds<BM>(As[p ^ 1], tid, ra);
      st_wtile_lds(Bs[p ^ 1], tid, rw);
    }
    __syncthreads();
    p ^= 1;
  }

  const float t = thr_p[0];
  const float s = step_p[0];

#pragma unroll
  for (int rt = 0; rt < 2; ++rt)
#pragma unroll
    for (int ct = 0; ct < 2; ++ct)
#pragma unroll
      for (int i = 0; i < 8; ++i) {
        const int row = b0 + wm + rt * 16 + hi * 8 + i;
        const int col = n0 + wn + ct * 16 + rc;
        const size_t idx = (size_t)row * NDIM + col;
        const float dv = haveD ? dprev[idx] : 0.0f;
        const float z  = dv - s * acc[rt][ct][i];
        const float az = fabsf(z) - t;
        out[idx] = az > 0.0f ? copysignf(az, z) : 0.0f;
      }
}

// ---------------------------------------------------------------------------
// Launch: 16 iterations x (rstep, zstep). r lives in d_ws (8 MB).
// Iterate t reads d from out[t-1]; t=0 uses haveD=0 (d == 0).
// ---------------------------------------------------------------------------
extern "C" void kernel_launch(void* const* d_in, const int* in_sizes, int n_in,
                              void* d_out, int out_size, void* d_ws, size_t ws_size,
                              hipStream_t stream) {
  const float* y    = (const float*)d_in[0];
  const float* A    = (const float*)d_in[1];
  const float* W    = (const float*)d_in[2];
  const float* thr  = (const float*)d_in[3];
  const float* step = (const float*)d_in[4];
  float* out = (float*)d_out;
  float* r   = (float*)d_ws;   // [BATCH, MDIM] scratch, 8 MB

  const dim3 blk(256);
  const dim3 g1(MDIM / BN, BATCH / BM);   // (8, 32)
  const dim3 g2(NDIM / BN, BATCH / BM);   // (32, 32)

  for (int t = 0; t < NITER; ++t) {
    const int haveD = (t > 0) ? 1 : 0;
    const float* dprev = haveD ? (out + (size_t)(t - 1) * BATCH * NDIM) : out;
    alista_rstep<<<g1, blk, 0, stream>>>(dprev, A, y, r, haveD);
    alista_zstep<<<g2, blk, 0, stream>>>(dprev, W, r,
                                         out + (size_t)t * BATCH * NDIM,
                                         thr + t, step + t, haveD);
  }
}